// Encoder_37426345017684
// MI455X (gfx1250) — compile-verified
//
#include <hip/hip_runtime.h>
#include <math.h>

#define NN   100000
#define EE   1600000
#define HH   64
#define FDIM 4
#define EPSN 1e-5f

typedef float v2f __attribute__((ext_vector_type(2)));
typedef float v8f __attribute__((ext_vector_type(8)));

// ---------------- utility ----------------
__global__ void fillf(float* p, float v, int n) {
    int i = blockIdx.x * blockDim.x + threadIdx.x;
    if (i < n) p[i] = v;
}

__global__ void deg_accum(const int* __restrict__ dst, float* __restrict__ deg) {
    int e = blockIdx.x * blockDim.x + threadIdx.x;
    if (e < EE) atomicAdd(&deg[dst[e]], 1.0f);
}

__global__ void dinv_kernel(float* deg) {
    int i = blockIdx.x * blockDim.x + threadIdx.x;
    if (i < NN) deg[i] = rsqrtf(deg[i] + 1.0f);
}

// ---------------- GraphNorm (single stats pass: sum + sumsq) ----------------
template <int D>
__global__ void colstats(const float* __restrict__ x, float* __restrict__ sums,
                         float* __restrict__ sumsq) {
    __shared__ float s1[256];
    __shared__ float s2[256];
    int tid = threadIdx.x;
    int f = tid % D;
    int r0 = tid / D;
    const int RPB = 256 / D;
    float a = 0.f, b = 0.f;
    for (int row = blockIdx.x * RPB + r0; row < NN; row += gridDim.x * RPB) {
        float v = x[(size_t)row * D + f];
        a += v; b += v * v;
    }
    s1[tid] = a; s2[tid] = b;
    __syncthreads();
    if (tid < D) {
        float ta = 0.f, tb = 0.f;
        #pragma unroll 4
        for (int j = 0; j < RPB; j++) { ta += s1[tid + j * D]; tb += s2[tid + j * D]; }
        atomicAdd(&sums[tid], ta);
        atomicAdd(&sumsq[tid], tb);
    }
}

__global__ void gnorm_apply(const float* __restrict__ x, float* __restrict__ y,
                            const float* __restrict__ sums, const float* __restrict__ sumsq,
                            const float* __restrict__ gw, const float* __restrict__ gb,
                            const float* __restrict__ gms, int D) {
    int i = blockIdx.x * blockDim.x + threadIdx.x;
    if (i >= NN * D) return;
    int f = i % D;
    float mean = sums[f] * (1.0f / NN);
    float ex2  = sumsq[f] * (1.0f / NN);
    float mm   = gms[f] * mean;
    // var of (x - ms*mean) = E[x^2] - 2*mm*E[x] + mm^2
    float var = ex2 - 2.0f * mm * mean + mm * mm;
    y[i] = gw[f] * (x[i] - mm) * rsqrtf(var + EPSN) + gb[f];
}

// ---------------- dense X@W via V_WMMA_F32_16X16X4_F32 ----------------
// One wave computes one 16x16 f32 tile of C = A(NNxK) @ B(Kx64).
// A 16x4 layout : lanes 0-15 M=0..15; v0 holds K = k0 + (lane>>4)*2, v1 holds K+1.
// B 4x16 layout : lanes 0-15 N=0..15; v0 holds K = k0 + (lane>>4)*2, v1 holds K+1.
// C 16x16 layout: VGPR i -> M = i + (lane>>4)*8, N = lane&15.
__global__ void gemm_wmma(const float* __restrict__ A, const float* __restrict__ B,
                          float* __restrict__ C, int K) {
    int wave = (blockIdx.x * blockDim.x + threadIdx.x) >> 5;
    int lane = threadIdx.x & 31;
    int tileM = wave >> 2;           // 64/16 = 4 tiles along N
    int tileN = wave & 3;
    int mrow = lane & 15;
    int khalf = lane >> 4;
    int ncol = lane & 15;
    const float* Arow = A + (size_t)(tileM * 16 + mrow) * K;
    const float* Bcol = B + tileN * 16 + ncol;
    v8f acc = {};
    for (int k0 = 0; k0 < K; k0 += 4) {
        int ka = k0 + khalf * 2;
        v2f a, b;
        a.x = Arow[ka];
        a.y = Arow[ka + 1];
        b.x = Bcol[(size_t)ka * HH];
        b.y = Bcol[(size_t)(ka + 1) * HH];
        acc = __builtin_amdgcn_wmma_f32_16x16x4_f32(
            /*neg_a=*/false, a, /*neg_b=*/false, b,
            /*c_mod=*/(short)0, acc, /*reuse_a=*/false, /*reuse_b=*/false);
    }
    int mbase = tileM * 16 + khalf * 8;
    int cc = tileN * 16 + ncol;
    #pragma unroll
    for (int i = 0; i < 8; i++)
        C[(size_t)(mbase + i) * HH + cc] = acc[i];
}

// ---------------- edge scatter (memory/atomic bound) ----------------
__global__ void edge_scatter(const float* __restrict__ xw, const int* __restrict__ src,
                             const int* __restrict__ dst, const float* __restrict__ dinv,
                             float* __restrict__ agg) {
    int idx = blockIdx.x * blockDim.x + threadIdx.x;
    if (idx >= EE * HH) return;
    int e = idx >> 6;           // / 64
    int f = idx & 63;
    int s = src[e], d = dst[e];
    float coef = dinv[s] * dinv[d];
    atomicAdd(&agg[(size_t)d * HH + f], xw[(size_t)s * HH + f] * coef);
}

__global__ void gcn_epilogue(float* __restrict__ agg, const float* __restrict__ xw,
                             const float* __restrict__ dinv, const float* __restrict__ bias,
                             const float* __restrict__ alpha, int use_prelu) {
    int i = blockIdx.x * blockDim.x + threadIdx.x;
    if (i >= NN * HH) return;
    int row = i >> 6, f = i & 63;
    float dv = dinv[row];
    float y = agg[i] + xw[i] * dv * dv + bias[f];
    if (use_prelu) y = (y >= 0.f) ? y : alpha[f] * y;
    agg[i] = y;
}

// ---------------- Set2Set ----------------
__device__ __forceinline__ unsigned encf(float f) {
    unsigned u = __float_as_uint(f);
    return (u & 0x80000000u) ? ~u : (u | 0x80000000u);
}
__device__ __forceinline__ float decf(unsigned u) {
    return __uint_as_float((u & 0x80000000u) ? (u & 0x7FFFFFFFu) : ~u);
}

__global__ void s2s_init(float* hs, float* c, float* rnum, float* rden) {
    int t = threadIdx.x;
    if (t < HH) { hs[t] = 0.f; c[t] = 0.f; rnum[t] = 0.f; }
    if (t == 0) rden[0] = 1.0f;   // q_star initial r = 0/1 = 0
}

__global__ void s2s_lstm(float* __restrict__ hs, float* __restrict__ c,
                         const float* __restrict__ rnum, const float* __restrict__ rden,
                         const float* __restrict__ Wih, const float* __restrict__ Whh,
                         const float* __restrict__ bih, const float* __restrict__ bhh) {
    __shared__ float q[2 * HH];
    __shared__ float gates[4 * HH];
    __shared__ float hsl[HH];
    int t = threadIdx.x;          // 256 threads, one per gate row
    if (t < HH)            { hsl[t] = hs[t]; q[t] = hsl[t]; }
    else if (t < 2 * HH)   { q[t] = rnum[t - HH] / rden[0]; }
    __syncthreads();
    float g = bih[t] + bhh[t];
    const float* wi = Wih + (size_t)t * (2 * HH);
    #pragma unroll 8
    for (int j = 0; j < 2 * HH; j++) g += q[j] * wi[j];
    const float* wh = Whh + (size_t)t * HH;
    #pragma unroll 8
    for (int j = 0; j < HH; j++) g += hsl[j] * wh[j];
    gates[t] = g;
    __syncthreads();
    if (t < HH) {
        float ig = 1.f / (1.f + expf(-gates[t]));
        float fg = 1.f / (1.f + expf(-gates[HH + t]));
        float gg = tanhf(gates[2 * HH + t]);
        float og = 1.f / (1.f + expf(-gates[3 * HH + t]));
        float cn = fg * c[t] + ig * gg;
        c[t] = cn;
        hs[t] = og * tanhf(cn);
    }
}

__global__ void s2s_reset(float* rnum, float* rden, unsigned* menc) {
    int t = threadIdx.x;
    if (t < HH) rnum[t] = 0.f;
    else if (t == HH) rden[0] = 0.f;
    else if (t == HH + 1) *menc = encf(-3.402823466e38f);
}

// wave per node: e[i] = dot(h[i], hs); running max via ordered-uint atomicMax
__global__ void s2s_logits(const float* __restrict__ h, const float* __restrict__ hs,
                           float* __restrict__ evec, unsigned* __restrict__ menc) {
    int wave = (blockIdx.x * blockDim.x + threadIdx.x) >> 5;
    int lane = threadIdx.x & 31;
    if (wave >= NN) return;
    const float* row = h + (size_t)wave * HH;
    float p = row[lane] * hs[lane] + row[lane + 32] * hs[lane + 32];
    #pragma unroll
    for (int off = 16; off > 0; off >>= 1) p += __shfl_down(p, off, 32);
    if (lane == 0) {
        evec[wave] = p;
        atomicMax(menc, encf(p));
    }
}

// rnum[f] += sum_i exp(e_i - m) * h[i][f];  rden += sum_i exp(e_i - m)
__global__ void s2s_accum(const float* __restrict__ h, const float* __restrict__ evec,
                          const unsigned* __restrict__ menc, float* __restrict__ rnum,
                          float* __restrict__ rden) {
    __shared__ float sm[256];
    __shared__ float sd[4];
    int t = threadIdx.x;
    int f = t & 63, sub = t >> 6;
    float m = decf(*menc);
    float acc = 0.f, accd = 0.f;
    for (int node = blockIdx.x * 4 + sub; node < NN; node += gridDim.x * 4) {
        float w = expf(evec[node] - m);
        acc += w * h[(size_t)node * HH + f];
        if (f == 0) accd += w;
    }
    sm[t] = acc;
    if (f == 0) sd[sub] = accd;
    __syncthreads();
    if (t < 64) atomicAdd(&rnum[t], sm[t] + sm[t + 64] + sm[t + 128] + sm[t + 192]);
    if (t == 0) atomicAdd(rden, sd[0] + sd[1] + sd[2] + sd[3]);
}

__global__ void s2s_final(const float* hs, const float* rnum, const float* rden,
                          float* out) {
    int t = threadIdx.x;
    if (t < HH) out[t] = hs[t];
    else if (t < 2 * HH) out[t] = rnum[t - HH] / rden[0];
}

// ---------------- host orchestration ----------------
static void run_gcn_layer(const float* in, int K, const float* Wm, const float* bias,
                          const float* alpha, const float* gw, const float* gb,
                          const float* gm, float* xw, float* agg, const float* dinv,
                          const int* src, const int* dst, float* sums, float* sumsq,
                          hipStream_t stream) {
    // X@W : 6250 * 4 tiles, 1 wave/tile, 8 waves/block
    gemm_wmma<<<3125, 256, 0, stream>>>(in, Wm, xw, K);
    fillf<<<(NN * HH + 255) / 256, 256, 0, stream>>>(agg, 0.f, NN * HH);
    edge_scatter<<<(EE * HH + 255) / 256, 256, 0, stream>>>(xw, src, dst, dinv, agg);
    gcn_epilogue<<<(NN * HH + 255) / 256, 256, 0, stream>>>(agg, xw, dinv, bias, alpha,
                                                            alpha != nullptr);
    fillf<<<1, 128, 0, stream>>>(sums, 0.f, 2 * HH);   // sums + sumsq contiguous
    colstats<HH><<<512, 256, 0, stream>>>(agg, sums, sumsq);
    gnorm_apply<<<(NN * HH + 255) / 256, 256, 0, stream>>>(agg, agg, sums, sumsq, gw, gb,
                                                           gm, HH);
}

extern "C" void kernel_launch(void* const* d_in, const int* in_sizes, int n_in,
                              void* d_out, int out_size, void* d_ws, size_t ws_size,
                              hipStream_t stream) {
    const float* x    = (const float*)d_in[0];
    const float* gn0w = (const float*)d_in[1];
    const float* gn0b = (const float*)d_in[2];
    const float* gn0m = (const float*)d_in[3];
    const float* W1   = (const float*)d_in[4];
    const float* b1   = (const float*)d_in[5];
    const float* a1   = (const float*)d_in[6];
    const float* gn1w = (const float*)d_in[7];
    const float* gn1b = (const float*)d_in[8];
    const float* gn1m = (const float*)d_in[9];
    const float* W2   = (const float*)d_in[10];
    const float* b2   = (const float*)d_in[11];
    const float* a2   = (const float*)d_in[12];
    const float* gn2w = (const float*)d_in[13];
    const float* gn2b = (const float*)d_in[14];
    const float* gn2m = (const float*)d_in[15];
    const float* W3   = (const float*)d_in[16];
    const float* b3   = (const float*)d_in[17];
    const float* gn3w = (const float*)d_in[18];
    const float* gn3b = (const float*)d_in[19];
    const float* gn3m = (const float*)d_in[20];
    const float* Wih  = (const float*)d_in[21];
    const float* Whh  = (const float*)d_in[22];
    const float* bih  = (const float*)d_in[23];
    const float* bhh  = (const float*)d_in[24];
    const int*   ei   = (const int*)d_in[25];
    const int* src = ei;
    const int* dst = ei + EE;

    float* W = (float*)d_ws;
    size_t o = 0;
    float* dinv  = W + o; o += NN;
    float* xn    = W + o; o += (size_t)NN * FDIM;
    float* bufA  = W + o; o += (size_t)NN * HH;
    float* bufB  = W + o; o += (size_t)NN * HH;   // xw scratch
    float* bufC  = W + o; o += (size_t)NN * HH;
    float* evec  = W + o; o += NN;
    float* sums  = W + o; o += HH;
    float* sumsq = W + o; o += HH;
    float* hs    = W + o; o += HH;
    float* cst   = W + o; o += HH;
    float* rnum  = W + o; o += HH;
    float* rden  = W + o; o += 1;
    unsigned* menc = (unsigned*)(W + o); o += 1;

    // degrees -> dinv = 1/sqrt(deg_in + 1)
    fillf<<<(NN + 255) / 256, 256, 0, stream>>>(dinv, 0.f, NN);
    deg_accum<<<(EE + 255) / 256, 256, 0, stream>>>(dst, dinv);
    dinv_kernel<<<(NN + 255) / 256, 256, 0, stream>>>(dinv);

    // GraphNorm on raw features (D=4)
    fillf<<<1, 128, 0, stream>>>(sums, 0.f, 2 * HH);
    colstats<FDIM><<<256, 256, 0, stream>>>(x, sums, sumsq);
    gnorm_apply<<<(NN * FDIM + 255) / 256, 256, 0, stream>>>(x, xn, sums, sumsq, gn0w,
                                                             gn0b, gn0m, FDIM);

    // 3 GCN layers (buffers rotate: out of layer k feeds layer k+1)
    run_gcn_layer(xn,   FDIM, W1, b1, a1,      gn1w, gn1b, gn1m, bufB, bufA, dinv, src, dst, sums, sumsq, stream);
    run_gcn_layer(bufA, HH,   W2, b2, a2,      gn2w, gn2b, gn2m, bufB, bufC, dinv, src, dst, sums, sumsq, stream);
    run_gcn_layer(bufC, HH,   W3, b3, nullptr, gn3w, gn3b, gn3m, bufB, bufA, dinv, src, dst, sums, sumsq, stream);
    float* hfin = bufA;

    // Set2Set (3 steps)
    s2s_init<<<1, 64, 0, stream>>>(hs, cst, rnum, rden);
    for (int s = 0; s < 3; s++) {
        s2s_lstm<<<1, 256, 0, stream>>>(hs, cst, rnum, rden, Wih, Whh, bih, bhh);
        s2s_reset<<<1, 128, 0, stream>>>(rnum, rden, menc);
        s2s_logits<<<(NN + 7) / 8, 256, 0, stream>>>(hfin, hs, evec, menc);
        s2s_accum<<<1024, 256, 0, stream>>>(hfin, evec, menc, rnum, rden);
    }
    s2s_final<<<1, 128, 0, stream>>>(hs, rnum, rden, (float*)d_out);
}